// GCN_89189290869192
// MI455X (gfx1250) — compile-verified
//
#include <hip/hip_runtime.h>
#include <hip/hip_bf16.h>

typedef __attribute__((ext_vector_type(2))) float v2f;
typedef __attribute__((ext_vector_type(8))) float v8f;

#define EPS 1e-5f

// ---------------- degree / norms ----------------
__global__ void k_deg_init(float* deg, int n) {
    int i = blockIdx.x * blockDim.x + threadIdx.x;
    if (i < n) deg[i] = 1.0f;              // self-loop
}
__global__ void k_deg_count(const int* __restrict__ dst, float* deg, int e) {
    int i = blockIdx.x * blockDim.x + threadIdx.x;
    if (i < e) atomicAdd(&deg[dst[i]], 1.0f);
}
__global__ void k_deg_rsqrt(float* deg, int n) {
    int i = blockIdx.x * blockDim.x + threadIdx.x;
    if (i < n) deg[i] = rsqrtf(deg[i]);    // deg now holds dinv
}

// ---------------- WMMA f32 GEMM: Y[n,Fout] = X[n,Fin] @ W[Fin,Fout] --------
// One wave32 computes a 16x16 tile of Y via V_WMMA_F32_16X16X4_F32.
// Fin/Fout are compile-time: K-loop fully unrolls, no divergent guards.
// For Fin not a multiple of 4, pad with clamped-index loads * {0,1} mask
// (branch-free, EXEC stays all-ones). n % 16 == 0, Fout % 16 == 0.
template<int Fin, int Fout>
__global__ __launch_bounds__(256) void k_gemm_wmma(const float* __restrict__ X,
                                                   const float* __restrict__ W,
                                                   float* __restrict__ Y, int n) {
    int wave = (int)((blockIdx.x * blockDim.x + threadIdx.x) >> 5);
    int lane = threadIdx.x & 31;
    constexpr int tilesN = Fout >> 4;
    int tilesM = n >> 4;
    if (wave >= tilesM * tilesN) return;
    int tm = wave / tilesN, tn = wave % tilesN;
    const float* xr = X + (size_t)(tm * 16 + (lane & 15)) * Fin;  // A row
    int col  = tn * 16 + (lane & 15);       // B/D column for this lane
    int kOff = (lane >> 4) << 1;            // 0 (lanes 0-15) or 2 (lanes 16-31)

    v8f acc = {};
#pragma unroll
    for (int k0 = 0; k0 < Fin; k0 += 4) {
        int ka = k0 + kOff;                 // in-bounds for all our Fin
        v2f a, b;
        if constexpr ((Fin & 3) == 0) {
            a.x = xr[ka];
            a.y = xr[ka + 1];
            b.x = W[(size_t)ka * Fout + col];
            b.y = W[(size_t)(ka + 1) * Fout + col];
        } else {
            // pad K to 4: clamp index, multiply by 0/1 mask (no branches)
            float m1  = (ka + 1 < Fin) ? 1.0f : 0.0f;
            int   ka1 = (ka + 1 < Fin) ? ka + 1 : Fin - 1;
            a.x = xr[ka];
            a.y = xr[ka1] * m1;
            b.x = W[(size_t)ka * Fout + col];
            b.y = W[(size_t)ka1 * Fout + col] * m1;
        }
        acc = __builtin_amdgcn_wmma_f32_16x16x4_f32(
            /*neg_a=*/false, a, /*neg_b=*/false, b,
            /*c_mod=*/(short)0, acc, /*reuse_a=*/false, /*reuse_b=*/false);
    }
    int rbase = tm * 16 + ((lane >> 4) << 3);   // +0 or +8
#pragma unroll
    for (int v = 0; v < 8; ++v)
        Y[(size_t)(rbase + v) * Fout + col] = acc[v];
}

// -------- agg init: self-loop term + bias:  agg = xw*dinv^2 + b ----------
__global__ void k_agg_init(const float* __restrict__ xw, const float* __restrict__ dinv,
                           const float* __restrict__ bias, float* __restrict__ agg,
                           int n, int F) {
    int t = blockIdx.x * blockDim.x + threadIdx.x;
    if (t >= n * F) return;
    int i = t / F, f = t % F;
    float d = dinv[i];
    agg[t] = xw[t] * d * d + bias[f];
}

// -------- edge scatter: agg[dst] += xw[src] * dinv[src]*dinv[dst] --------
__global__ void k_edge_scatter(const int* __restrict__ src, const int* __restrict__ dst,
                               const float* __restrict__ xw, const float* __restrict__ dinv,
                               float* __restrict__ agg, int e, int F) {
    int t = blockIdx.x * blockDim.x + threadIdx.x;
    if (t >= e * F) return;
    int ed = t / F, f = t % F;
    int s = src[ed], d = dst[ed];
    float c = dinv[s] * dinv[d];
    atomicAdd(&agg[(size_t)d * F + f], xw[(size_t)s * F + f] * c);
}

// -------- batchnorm stats: one block per feature --------
__global__ void k_bn_stats(const float* __restrict__ A, float* __restrict__ mean,
                           float* __restrict__ var, int n, int F) {
    int f = blockIdx.x;
    __shared__ float s1[256], s2[256];
    float sum = 0.f, sum2 = 0.f;
    for (int i = threadIdx.x; i < n; i += blockDim.x) {
        float v = A[(size_t)i * F + f];
        sum += v; sum2 += v * v;
    }
    s1[threadIdx.x] = sum; s2[threadIdx.x] = sum2;
    __syncthreads();
    for (int s = 128; s > 0; s >>= 1) {
        if (threadIdx.x < s) { s1[threadIdx.x] += s1[threadIdx.x + s];
                               s2[threadIdx.x] += s2[threadIdx.x + s]; }
        __syncthreads();
    }
    if (threadIdx.x == 0) {
        float m = s1[0] / (float)n;
        mean[f] = m;
        var[f]  = s2[0] / (float)n - m * m;
    }
}

// -------- batchnorm apply + relu --------
__global__ void k_bn_relu(const float* __restrict__ A, const float* __restrict__ mean,
                          const float* __restrict__ var, const float* __restrict__ g,
                          const float* __restrict__ be, float* __restrict__ H,
                          int n, int F) {
    int t = blockIdx.x * blockDim.x + threadIdx.x;
    if (t >= n * F) return;
    int f = t % F;
    float v = g[f] * (A[t] - mean[f]) * rsqrtf(var[f] + EPS) + be[f];
    H[t] = v > 0.f ? v : 0.f;
}

// -------- segment-max pooling (post-ReLU values >= 0 -> int punning OK) ----
__global__ void k_pool_init(float* pooled, int gf) {
    int t = blockIdx.x * blockDim.x + threadIdx.x;
    if (t < gf) pooled[t] = 0.0f;
}
__global__ void k_pool_max(const float* __restrict__ H, const int* __restrict__ batch,
                           float* __restrict__ pooled, int n, int F) {
    int t = blockIdx.x * blockDim.x + threadIdx.x;
    if (t >= n * F) return;
    int i = t / F, f = t % F;
    int gidx = batch[i];
    atomicMax((int*)&pooled[(size_t)gidx * F + f], __float_as_int(H[t]));
}

// -------- FC 64->10 + softmax; out = [logits(G*10) | softmax(G*10)] --------
__global__ void k_fc_softmax(const float* __restrict__ pooled, const float* __restrict__ W,
                             const float* __restrict__ b, float* __restrict__ out, int G) {
    int g = blockIdx.x;
    int c = threadIdx.x;
    __shared__ float sl[10], se[10];
    float logit = 0.f;
    if (c < 10) {
        float acc = b[c];
        for (int k = 0; k < 64; ++k)
            acc += pooled[(size_t)g * 64 + k] * W[(size_t)k * 10 + c];
        logit = acc;
        sl[c] = acc;
        out[(size_t)g * 10 + c] = acc;
    }
    __syncthreads();
    float m = sl[0], e = 0.f;
    if (c < 10) {
        for (int k = 1; k < 10; ++k) m = sl[k] > m ? sl[k] : m;
        e = __expf(logit - m);
        se[c] = e;
    }
    __syncthreads();
    if (c < 10) {
        float s = 0.f;
        for (int k = 0; k < 10; ++k) s += se[k];
        out[(size_t)G * 10 + (size_t)g * 10 + c] = e / s;
    }
}

extern "C" void kernel_launch(void* const* d_in, const int* in_sizes, int n_in,
                              void* d_out, int out_size, void* d_ws, size_t ws_size,
                              hipStream_t stream) {
    const float* x     = (const float*)d_in[0];
    const int*   ei    = (const int*)  d_in[1];
    const int*   batch = (const int*)  d_in[2];
    const float* Wl[3] = { (const float*)d_in[3],  (const float*)d_in[7],  (const float*)d_in[11] };
    const float* bl[3] = { (const float*)d_in[4],  (const float*)d_in[8],  (const float*)d_in[12] };
    const float* gl[3] = { (const float*)d_in[5],  (const float*)d_in[9],  (const float*)d_in[13] };
    const float* el[3] = { (const float*)d_in[6],  (const float*)d_in[10], (const float*)d_in[14] };
    const float* fcw   = (const float*)d_in[15];
    const float* fcb   = (const float*)d_in[16];

    const int N = in_sizes[0] / 3;
    const int E = in_sizes[1] / 2;
    const int G = out_size / 20;            // logits + softmax, 10 classes each
    const int* src = ei;
    const int* dst = ei + E;

    // workspace layout (floats)
    float* ws     = (float*)d_ws;
    float* dinv   = ws;                         // N  (degree -> dinv in place)
    float* xw     = dinv + N;                   // N*64
    float* agg    = xw  + (size_t)N * 64;       // N*64
    float* h      = agg + (size_t)N * 64;       // N*64
    float* meanb  = h   + (size_t)N * 64;       // 64
    float* varb   = meanb + 64;                 // 64
    float* pooled = varb  + 64;                 // G*64

    const int B = 256;
    auto cdiv = [](long long a, long long b) { return (int)((a + b - 1) / b); };

    // degrees (same for all layers)
    k_deg_init <<<cdiv(N, B), B, 0, stream>>>(dinv, N);
    k_deg_count<<<cdiv(E, B), B, 0, stream>>>(dst, dinv, E);
    k_deg_rsqrt<<<cdiv(N, B), B, 0, stream>>>(dinv, N);

    const int Fin_[3]  = { 3, 16, 32 };
    const int Fout_[3] = { 16, 32, 64 };
    const float* in = x;

    for (int l = 0; l < 3; ++l) {
        int Fin = Fin_[l], Fout = Fout_[l];
        // xw = in @ W  (WMMA f32 16x16x4 tiles, compile-time shapes)
        long long waveThreads = (long long)(N >> 4) * (Fout >> 4) * 32;
        if (l == 0)
            k_gemm_wmma<3, 16> <<<cdiv(waveThreads, B), B, 0, stream>>>(in, Wl[l], xw, N);
        else if (l == 1)
            k_gemm_wmma<16, 32><<<cdiv(waveThreads, B), B, 0, stream>>>(in, Wl[l], xw, N);
        else
            k_gemm_wmma<32, 64><<<cdiv(waveThreads, B), B, 0, stream>>>(in, Wl[l], xw, N);
        // agg = xw*dinv^2 + b, then edge scatter-add
        k_agg_init<<<cdiv((long long)N * Fout, B), B, 0, stream>>>(xw, dinv, bl[l], agg, N, Fout);
        k_edge_scatter<<<cdiv((long long)E * Fout, B), B, 0, stream>>>(src, dst, xw, dinv, agg, E, Fout);
        // batchnorm + relu -> h
        k_bn_stats<<<Fout, B, 0, stream>>>(agg, meanb, varb, N, Fout);
        k_bn_relu<<<cdiv((long long)N * Fout, B), B, 0, stream>>>(agg, meanb, varb, gl[l], el[l], h, N, Fout);
        in = h;
    }

    // segment-max pool over batch, then FC + softmax
    k_pool_init<<<cdiv((long long)G * 64, B), B, 0, stream>>>(pooled, G * 64);
    k_pool_max <<<cdiv((long long)N * 64, B), B, 0, stream>>>(h, batch, pooled, N, 64);
    k_fc_softmax<<<G, 32, 0, stream>>>(pooled, fcw, fcb, (float*)d_out, G);
}